// SlidingWindowMultiHeadSelfAttention_4698694221953
// MI455X (gfx1250) — compile-verified
//
#include <hip/hip_runtime.h>
#include <hip/hip_bf16.h>

// ---------------------------------------------------------------------------
// Sliding-window MHSA forward for gfx1250 (MI455X), fp32 via V_WMMA_F32_16X16X4_F32
// B=4, S=2048, E=768, H=12, HD=64
// ---------------------------------------------------------------------------

typedef __attribute__((ext_vector_type(2))) float v2f;
typedef __attribute__((ext_vector_type(8))) float v8f;

#define BB 4
#define SS 2048
#define EE 768
#define HH 12
#define HD 64
#define PADV (-1e9f)
#define KCAP 448   // max key span per 16-query tile: 15 + 2*205 + align < 448

__device__ __forceinline__ v8f wmma4(v2f a, v2f b, v8f c) {
  // D = A(16x4 f32) * B(4x16 f32) + C(16x16 f32); full fp32 precision
  return __builtin_amdgcn_wmma_f32_16x16x4_f32(false, a, false, b, (short)0, c,
                                               false, false);
}

// ---------------------------------------------------------------------------
// Kernel 1: per-batch window size = max(ceil(valid_len * 10/100), 2)
// ---------------------------------------------------------------------------
__global__ __launch_bounds__(256) void wsz_kernel(const int* __restrict__ mask,
                                                  int* __restrict__ wsz) {
  __shared__ int red[256];
  const int b = blockIdx.x;
  int cnt = 0;
  for (int s = threadIdx.x; s < SS; s += 256) cnt += (mask[b * SS + s] == 0);
  red[threadIdx.x] = cnt;
  __syncthreads();
  for (int off = 128; off; off >>= 1) {
    if ((int)threadIdx.x < off) red[threadIdx.x] += red[threadIdx.x + off];
    __syncthreads();
  }
  if (threadIdx.x == 0) {
    int len = red[0];
    if (len > SS) len = SS;
    int w = (len * 10 + 99) / 100;  // ceil(len*0.1)
    if (w < 2) w = 2;
    wsz[b] = w;
  }
}

// ---------------------------------------------------------------------------
// Wave-level GEMM core: C(16 x NT*16) = A[m0:,:] @ B[n0:,:]^T
// A row-major [M,K], B row-major [N,K]. Fragments are contiguous v2f loads.
// ---------------------------------------------------------------------------
template <int NT>
__device__ __forceinline__ void wave_gemm(const float* __restrict__ A, int lda,
                                          const float* __restrict__ Bm, int ldb,
                                          int m0, int n0, int K, v8f* acc) {
  const int lane = threadIdx.x & 31;
  const int half = lane >> 4;
  const int r = lane & 15;
  const float* arow = A + (size_t)(m0 + r) * lda + 2 * half;
  const float* brow[NT];
#pragma unroll
  for (int t = 0; t < NT; ++t)
    brow[t] = Bm + (size_t)(n0 + t * 16 + r) * ldb + 2 * half;
  for (int k = 0; k < K; k += 4) {
    v2f a = *(const v2f*)(arow + k);
#pragma unroll
    for (int t = 0; t < NT; ++t) {
      v2f bf = *(const v2f*)(brow[t] + k);
      acc[t] = wmma4(a, bf, acc[t]);
    }
  }
}

// ---------------------------------------------------------------------------
// Kernel 2: QKV projection. x[8192,768] @ qkv_w[2304,768]^T + qkv_b.
// Scatter: q,k -> [B,H,S,HD]; v -> TRANSPOSED [B,H,HD,S] (for v2f PV frags).
// grid (512, 9), block 128 (4 waves, each wave: 16x64 tile, NT=4)
// ---------------------------------------------------------------------------
__global__ __launch_bounds__(128) void qkv_gemm(const float* __restrict__ x,
                                                const float* __restrict__ qkv_w,
                                                const float* __restrict__ qkv_b,
                                                float* __restrict__ wq,
                                                float* __restrict__ wk,
                                                float* __restrict__ wv) {
  const int wave = threadIdx.x >> 5;
  const int lane = threadIdx.x & 31;
  const int half = lane >> 4;
  const int r = lane & 15;
  const int m0 = blockIdx.x * 16;
  const int n0 = blockIdx.y * 256 + wave * 64;

  v8f acc[4] = {};
  wave_gemm<4>(x, EE, qkv_w, EE, m0, n0, EE, acc);

#pragma unroll
  for (int t = 0; t < 4; ++t) {
    const int n = n0 + t * 16 + r;
    const int h = n / (3 * HD);
    const int rem = n - h * (3 * HD);
    const int c = rem >> 6;  // 0=q 1=k 2=v  (wave-uniform per t)
    const int d = rem & 63;
    const float bias = qkv_b[n];
    if (c == 2) {  // V transposed: [B,H,HD,S]
#pragma unroll
      for (int i = 0; i < 8; ++i) {
        const int m = m0 + i + 8 * half;
        const int bi = m >> 11;
        const int s = m & (SS - 1);
        wv[(((size_t)(bi * HH + h)) * HD + d) * SS + s] = acc[t][i] + bias;
      }
    } else {
      float* dst = (c == 0) ? wq : wk;
#pragma unroll
      for (int i = 0; i < 8; ++i) {
        const int m = m0 + i + 8 * half;
        const int bi = m >> 11;
        const int s = m & (SS - 1);
        dst[(((size_t)(bi * HH + h)) * SS + s) * HD + d] = acc[t][i] + bias;
      }
    }
  }
}

// ---------------------------------------------------------------------------
// Kernel 3: zero attn_mean region (out-of-window probs are exactly 0)
// ---------------------------------------------------------------------------
__global__ __launch_bounds__(256) void zero_attn(float4* __restrict__ p, size_t n4) {
  size_t i = (size_t)blockIdx.x * blockDim.x + threadIdx.x;
  const size_t stride = (size_t)gridDim.x * blockDim.x;
  const float4 z = {0.f, 0.f, 0.f, 0.f};
  for (; i < n4; i += stride) p[i] = z;
}

// ---------------------------------------------------------------------------
// Kernel 4: windowed attention. One block per (b, 16-query tile).
// Per head: QK^T (WMMA) -> LDS logits -> mask+softmax -> P@V (WMMA, split-K
// across 8 waves, ds_add_f32 reduction) -> vbuf. Head-mean attn kept in LDS.
// ---------------------------------------------------------------------------
__global__ __launch_bounds__(256) void attn_kernel(
    const float* __restrict__ wq, const float* __restrict__ wk,
    const float* __restrict__ wv, const float* __restrict__ matrix_mask,
    const int* __restrict__ wsz, float* __restrict__ vbuf,
    float* __restrict__ attn_mean) {
  __shared__ float slog[16][KCAP];   // logits -> probabilities (28 KB)
  __shared__ float ssum[16][KCAP];   // sum over heads          (28 KB)
  __shared__ float sout[16][HD];     // PV split-K reduction     (4 KB)
  __shared__ float srowmax[16];
  __shared__ float srowsum[16];

  const int tid = threadIdx.x;
  const int wave = tid >> 5;
  const int lane = tid & 31;
  const int half = lane >> 4;
  const int r = lane & 15;

  const int b = blockIdx.y;
  const int q0 = blockIdx.x * 16;
  const int w = wsz[b];

  int klo = q0 - w;
  if (klo < 0) klo = 0;
  klo &= ~15;
  int khi = q0 + 15 + w;
  if (khi > SS - 1) khi = SS - 1;
  int nkt = ((khi - klo) >> 4) + 1;
  if (nkt > KCAP / 16) nkt = KCAP / 16;
  const int nk = nkt * 16;

  for (int idx = tid; idx < 16 * KCAP; idx += 256) (&ssum[0][0])[idx] = 0.f;
  __syncthreads();

  for (int h = 0; h < HH; ++h) {
    const float* qh = wq + ((size_t)(b * HH + h)) * SS * HD;
    const float* kh = wk + ((size_t)(b * HH + h)) * SS * HD;
    const float* vh = wv + ((size_t)(b * HH + h)) * HD * SS;  // transposed

    // Preload this tile's Q fragments (16 x 64, v2f per K-step per lane)
    v2f qf[16];
#pragma unroll
    for (int kd = 0; kd < 16; ++kd)
      qf[kd] = *(const v2f*)(qh + (size_t)(q0 + r) * HD + kd * 4 + 2 * half);

    // ---- QK^T over key tiles (wave-uniform loop; EXEC stays full) ----
    for (int kt = wave; kt < nkt; kt += 8) {
      v8f acc = {};
      const float* krow = kh + (size_t)(klo + kt * 16 + r) * HD + 2 * half;
#pragma unroll
      for (int kd = 0; kd < 16; ++kd) {
        v2f bf = *(const v2f*)(krow + kd * 4);
        acc = wmma4(qf[kd], bf, acc);
      }
#pragma unroll
      for (int i = 0; i < 8; ++i) {
        const int row = i + 8 * half;
        const int q = q0 + row;
        const int col = kt * 16 + r;
        const int key = klo + col;
        const float mm = matrix_mask[((size_t)b * SS + q) * SS + key];
        int dist = q - key;
        if (dist < 0) dist = -dist;
        const float mval = (dist > w || mm == PADV) ? PADV : mm;
        slog[row][col] = acc[i] * 0.125f + mval;  // 1/sqrt(64)
      }
    }
    __syncthreads();

    // ---- softmax row max / sum (16 threads per row, shfl width-16) ----
    {
      const int row = tid >> 4, sub = tid & 15;
      float mx = -3.4e38f;
      for (int c = sub; c < nk; c += 16) {
        const float v = slog[row][c];
        mx = v > mx ? v : mx;
      }
      for (int off = 8; off; off >>= 1) {
        const float o = __shfl_xor(mx, off, 16);
        mx = o > mx ? o : mx;
      }
      float sm = 0.f;
      for (int c = sub; c < nk; c += 16) sm += __expf(slog[row][c] - mx);
      for (int off = 8; off; off >>= 1) sm += __shfl_xor(sm, off, 16);
      if (sub == 0) {
        srowmax[row] = mx;
        srowsum[row] = sm;
      }
    }
    __syncthreads();

    // ---- logits -> probabilities, accumulate head sum, zero sout ----
    for (int idx = tid; idx < 16 * nk; idx += 256) {
      const int row = idx / nk, c = idx - row * nk;
      const float p = __expf(slog[row][c] - srowmax[row]) / srowsum[row];
      slog[row][c] = p;
      ssum[row][c] += p;
    }
    for (int idx = tid; idx < 16 * HD; idx += 256) (&sout[0][0])[idx] = 0.f;
    __syncthreads();

    // ---- P @ V : 8 waves = 4 d-tiles x 2 K-halves, LDS-atomic reduce ----
    {
      const int d0 = (wave & 3) * 16;
      const int khN = nk >> 1;  // multiple of 8
      const int kbeg = (wave >> 2) * khN;
      const int kend = kbeg + khN;
      const float* vrow = vh + (size_t)(d0 + r) * SS + klo + 2 * half;
      v8f acc = {};
      for (int kk = kbeg; kk < kend; kk += 4) {
        v2f a = *(const v2f*)(&slog[r][kk + 2 * half]);
        v2f bf = *(const v2f*)(vrow + kk);  // V^T: contiguous b64 load
        acc = wmma4(a, bf, acc);
      }
#pragma unroll
      for (int i = 0; i < 8; ++i)
        atomicAdd(&sout[i + 8 * half][d0 + r], acc[i]);  // ds_add_f32
    }
    __syncthreads();

    // ---- store per-head values into vbuf [B,S,E] ----
    for (int idx = tid; idx < 16 * HD; idx += 256) {
      const int row = idx >> 6, d = idx & 63;
      vbuf[((size_t)(b * SS + q0 + row)) * EE + h * HD + d] = sout[row][d];
    }
    __syncthreads();
  }

  // ---- attn_mean = ssum / H (in-window span; rest pre-zeroed) ----
  for (int idx = tid; idx < 16 * nk; idx += 256) {
    const int row = idx / nk, c = idx - row * nk;
    attn_mean[((size_t)b * SS + (q0 + row)) * SS + klo + c] =
        ssum[row][c] * (1.f / (float)HH);
  }
}

// ---------------------------------------------------------------------------
// Kernel 5: output projection. vbuf[8192,768] @ o_w[768,768]^T + o_b
// grid (512, 3), block 128 (4 waves, 16x64 per wave)
// ---------------------------------------------------------------------------
__global__ __launch_bounds__(128) void out_gemm(const float* __restrict__ vbuf,
                                                const float* __restrict__ ow,
                                                const float* __restrict__ ob,
                                                float* __restrict__ out) {
  const int wave = threadIdx.x >> 5;
  const int lane = threadIdx.x & 31;
  const int half = lane >> 4;
  const int r = lane & 15;
  const int m0 = blockIdx.x * 16;
  const int n0 = blockIdx.y * 256 + wave * 64;

  v8f acc[4] = {};
  wave_gemm<4>(vbuf, EE, ow, EE, m0, n0, EE, acc);

#pragma unroll
  for (int t = 0; t < 4; ++t) {
    const int n = n0 + t * 16 + r;
    const float bias = ob[n];
#pragma unroll
    for (int i = 0; i < 8; ++i) {
      const int m = m0 + i + 8 * half;
      out[(size_t)m * EE + n] = acc[t][i] + bias;
    }
  }
}

// ---------------------------------------------------------------------------
extern "C" void kernel_launch(void* const* d_in, const int* in_sizes, int n_in,
                              void* d_out, int out_size, void* d_ws,
                              size_t ws_size, hipStream_t stream) {
  const float* x = (const float*)d_in[0];
  const int* pad = (const int*)d_in[1];
  const float* mmask = (const float*)d_in[2];
  const float* qkv_w = (const float*)d_in[3];
  const float* qkv_b = (const float*)d_in[4];
  const float* o_w = (const float*)d_in[5];
  const float* o_b = (const float*)d_in[6];

  float* out = (float*)d_out;
  float* o = out;                                   // [B,S,E]
  float* attn_mean = out + (size_t)BB * SS * EE;    // [B,S,S]

  float* ws = (float*)d_ws;
  const size_t QSZ = (size_t)BB * HH * SS * HD;     // 6,291,456 floats
  float* wq = ws;
  float* wk = ws + QSZ;
  float* wv = ws + 2 * QSZ;                         // [B,H,HD,S] transposed
  float* vbuf = ws + 3 * QSZ;                       // [B,S,E]
  int* wsz = (int*)(ws + 4 * QSZ);

  wsz_kernel<<<BB, 256, 0, stream>>>(pad, wsz);
  qkv_gemm<<<dim3((BB * SS) / 16, (3 * EE) / 256), 128, 0, stream>>>(
      x, qkv_w, qkv_b, wq, wk, wv);
  zero_attn<<<4096, 256, 0, stream>>>((float4*)attn_mean,
                                      (size_t)BB * SS * SS / 4);
  attn_kernel<<<dim3(SS / 16, BB), 256, 0, stream>>>(wq, wk, wv, mmask, wsz,
                                                     vbuf, attn_mean);
  out_gemm<<<dim3((BB * SS) / 16, EE / 256), 128, 0, stream>>>(vbuf, o_w, o_b,
                                                               o);
}